// DualTowerGCN_40965398069665
// MI455X (gfx1250) — compile-verified
//
#include <hip/hip_runtime.h>
#include <hip/hip_bf16.h>

typedef __attribute__((ext_vector_type(2))) float v2f;
typedef __attribute__((ext_vector_type(8))) float v8f;

#define OUTDIM 128

// ---------------- degree / norm ----------------
__global__ void deg_init_kernel(float* deg, int n) {
    int i = blockIdx.x * blockDim.x + threadIdx.x;
    if (i < n) deg[i] = 1.0f;  // self-loop weight
}

__global__ void deg_accum_kernel(const int* __restrict__ col,
                                 const float* __restrict__ w,
                                 float* deg, int e) {
    int i = blockIdx.x * blockDim.x + threadIdx.x;
    if (i < e) atomicAdd(&deg[col[i]], w[i]);
}

__global__ void deg_rsqrt_kernel(float* deg, int n) {
    int i = blockIdx.x * blockDim.x + threadIdx.x;
    if (i < n) {
        float d = deg[i];
        deg[i] = (d > 0.0f) ? rsqrtf(d) : 0.0f;
    }
}

// ---------------- dense GEMM via f32 WMMA ----------------
// H[M x 128] = X[M x K] @ W[K x 128].  One wave per 16x16 output tile.
// A frag (16x4 f32, 2 VGPR): lanes 0-15 hold K=k0,k0+1 ; lanes 16-31 hold K=k0+2,k0+3
// B frag (4x16 f32, 2 VGPR): symmetric (N = lane%16)
// C/D (16x16 f32, 8 VGPR): VGPR r -> M = r + 8*(lane/16), N = lane%16
__global__ void gemm_wmma_f32_kernel(const float* __restrict__ X,
                                     const float* __restrict__ W,
                                     float* __restrict__ H,
                                     int M, int K) {
    int wave = (int)((blockIdx.x * blockDim.x + threadIdx.x) >> 5);
    int lane = threadIdx.x & 31;
    const int ntN = OUTDIM / 16;          // 8 tiles along N
    int mt = wave / ntN;
    int nt = wave % ntN;
    if (mt * 16 >= M) return;             // wave-uniform: EXEC stays all-1s for WMMA

    int h = lane >> 4;                    // 0 or 1 (half-wave)
    int r = lane & 15;

    const float* xrow = X + (size_t)(mt * 16 + r) * (size_t)K;
    const float* wcol = W + nt * 16 + r;

    v8f c = {};
    for (int k0 = 0; k0 < K; k0 += 4) {
        int ka = k0 + 2 * h;
        v2f a;
        a.x = xrow[ka];
        a.y = xrow[ka + 1];
        v2f b;
        b.x = wcol[(size_t)ka * OUTDIM];
        b.y = wcol[(size_t)(ka + 1) * OUTDIM];
        c = __builtin_amdgcn_wmma_f32_16x16x4_f32(
                /*neg_a=*/false, a, /*neg_b=*/false, b,
                /*c_mod=*/(short)0, c, /*reuse_a=*/false, /*reuse_b=*/false);
    }

    float* hout = H + (size_t)(mt * 16 + 8 * h) * OUTDIM + nt * 16 + r;
    #pragma unroll
    for (int i = 0; i < 8; ++i)
        hout[(size_t)i * OUTDIM] = c[i];
}

// ---------------- self-loop contribution + bias (initializes agg buffer) ----
__global__ void self_loop_bias_kernel(const float* __restrict__ H,
                                      const float* __restrict__ dinv,
                                      const float* __restrict__ bias,
                                      float* __restrict__ out, int n) {
    long long t = (long long)blockIdx.x * blockDim.x + threadIdx.x;
    long long total = (long long)n * OUTDIM;
    if (t >= total) return;
    int i = (int)(t >> 7);
    int ch = (int)(t & (OUTDIM - 1));
    float d = dinv[i];
    out[t] = d * d * H[t] + bias[ch];
}

// ---------------- edge scatter: out[col] += dinv[row]*w*dinv[col] * h[row] --
__global__ void scatter_edges_kernel(const float* __restrict__ H,
                                     const int* __restrict__ rowi,
                                     const int* __restrict__ coli,
                                     const float* __restrict__ w,
                                     const float* __restrict__ dinv,
                                     float* out, int e) {
    long long t = (long long)blockIdx.x * blockDim.x + threadIdx.x;
    int edge = (int)(t >> 5);
    int lane = (int)(t & 31);
    if (edge >= e) return;
    int rsrc = rowi[edge];
    int cdst = coli[edge];
    float norm = dinv[rsrc] * w[edge] * dinv[cdst];
    const float4 hv = ((const float4*)(H + (size_t)rsrc * OUTDIM))[lane];
    float* o = out + (size_t)cdst * OUTDIM + lane * 4;
    atomicAdd(o + 0, norm * hv.x);
    atomicAdd(o + 1, norm * hv.y);
    atomicAdd(o + 2, norm * hv.z);
    atomicAdd(o + 3, norm * hv.w);
}

// -------- pair max-pool + FC(128->1) + sigmoid, one wave per cluster -------
__global__ void pool_fc_kernel(const float* __restrict__ X,
                               const float* __restrict__ Wfc,
                               const float* __restrict__ bfc,
                               float* __restrict__ out,
                               int nclusters, int out_offset) {
    int wave = (int)((blockIdx.x * blockDim.x + threadIdx.x) >> 5);
    int lane = threadIdx.x & 31;
    if (wave >= nclusters) return;
    const float4 a = ((const float4*)(X + (size_t)(2 * wave) * OUTDIM))[lane];
    const float4 b = ((const float4*)(X + (size_t)(2 * wave + 1) * OUTDIM))[lane];
    const float4 wv = ((const float4*)Wfc)[lane];
    float s = fmaxf(a.x, b.x) * wv.x + fmaxf(a.y, b.y) * wv.y +
              fmaxf(a.z, b.z) * wv.z + fmaxf(a.w, b.w) * wv.w;
    #pragma unroll
    for (int ofs = 16; ofs > 0; ofs >>= 1)
        s += __shfl_down(s, ofs, 32);
    if (lane == 0)
        out[out_offset + wave] = 1.0f / (1.0f + expf(-(s + bfc[0])));
}

// ---------------------------------------------------------------------------
static inline int cdiv(long long a, int b) { return (int)((a + b - 1) / b); }

extern "C" void kernel_launch(void* const* d_in, const int* in_sizes, int n_in,
                              void* d_out, int out_size, void* d_ws, size_t ws_size,
                              hipStream_t stream) {
    (void)n_in; (void)out_size; (void)ws_size;

    const float* x1  = (const float*)d_in[0];
    const int*   ei1 = (const int*)d_in[1];
    const float* ew1 = (const float*)d_in[2];
    const float* x2  = (const float*)d_in[3];
    const int*   ei2 = (const int*)d_in[4];
    const float* ew2 = (const float*)d_in[5];
    // d_in[6..9]: cluster1/2, batch1/2 — clusters are arange//2, handled implicitly
    const float* W1a = (const float*)d_in[10];
    const float* b1a = (const float*)d_in[11];
    const float* W1b = (const float*)d_in[12];
    const float* b1b = (const float*)d_in[13];
    const float* W2a = (const float*)d_in[14];
    const float* b2a = (const float*)d_in[15];
    const float* W2b = (const float*)d_in[16];
    const float* b2b = (const float*)d_in[17];
    const float* Wfc = (const float*)d_in[18];
    const float* bfc = (const float*)d_in[19];

    const int E1 = in_sizes[2];
    const int E2 = in_sizes[5];
    const int N1 = in_sizes[0] / 128;      // IN1 = 128
    const int N2 = in_sizes[3] / 64;       // IN2 = 64
    const int C1 = N1 / 2;
    const int C2 = N2 / 2;
    const int Nmax = (N1 > N2) ? N1 : N2;

    // workspace layout: bufH | bufO | deg
    float* bufH = (float*)d_ws;
    float* bufO = bufH + (size_t)Nmax * OUTDIM;
    float* deg  = bufO + (size_t)Nmax * OUTDIM;

    const int BT = 256;

    struct Tower {
        const float* x; int K; const int* ei; const float* ew;
        const float* Wa; const float* ba; const float* Wb; const float* bb;
        int N; int E; int C; int out_off;
    } towers[2] = {
        { x1, 128, ei1, ew1, W1a, b1a, W1b, b1b, N1, E1, C1, 0  },
        { x2,  64, ei2, ew2, W2a, b2a, W2b, b2b, N2, E2, C2, C1 },
    };

    for (int t = 0; t < 2; ++t) {
        const Tower& tw = towers[t];
        const int* rowp = tw.ei;            // edge_index[0]
        const int* colp = tw.ei + tw.E;     // edge_index[1]

        // normalization: deg -> dinv (in place)
        deg_init_kernel<<<cdiv(tw.N, BT), BT, 0, stream>>>(deg, tw.N);
        deg_accum_kernel<<<cdiv(tw.E, BT), BT, 0, stream>>>(colp, tw.ew, deg, tw.E);
        deg_rsqrt_kernel<<<cdiv(tw.N, BT), BT, 0, stream>>>(deg, tw.N);

        // ---- conv A: h = x @ Wa ; agg = scatter(norm*h) + ba ----
        {
            long long waves = (long long)cdiv(tw.N, 16) * (OUTDIM / 16);
            gemm_wmma_f32_kernel<<<cdiv(waves * 32, BT), BT, 0, stream>>>(
                tw.x, tw.Wa, bufH, tw.N, tw.K);
        }
        self_loop_bias_kernel<<<cdiv((long long)tw.N * OUTDIM, BT), BT, 0, stream>>>(
            bufH, deg, tw.ba, bufO, tw.N);
        scatter_edges_kernel<<<cdiv((long long)tw.E * 32, BT), BT, 0, stream>>>(
            bufH, rowp, colp, tw.ew, deg, bufO, tw.E);

        // ---- conv B: h = agg @ Wb ; agg = scatter(norm*h) + bb ----
        {
            long long waves = (long long)cdiv(tw.N, 16) * (OUTDIM / 16);
            gemm_wmma_f32_kernel<<<cdiv(waves * 32, BT), BT, 0, stream>>>(
                bufO, tw.Wb, bufH, tw.N, OUTDIM);
        }
        self_loop_bias_kernel<<<cdiv((long long)tw.N * OUTDIM, BT), BT, 0, stream>>>(
            bufH, deg, tw.bb, bufO, tw.N);
        // NOTE: scatter writes into bufO which self_loop_bias just initialized;
        // gemm consumed the previous bufO before this point (stream ordered).
        scatter_edges_kernel<<<cdiv((long long)tw.E * 32, BT), BT, 0, stream>>>(
            bufH, rowp, colp, tw.ew, deg, bufO, tw.E);

        // wait: scatter above reads bufH, writes bufO — but bufH was the gemm
        // output and bufO the conv-B aggregation target. Correct.

        // ---- pair max-pool + FC + sigmoid ----
        pool_fc_kernel<<<cdiv((long long)tw.C * 32, BT), BT, 0, stream>>>(
            bufO, Wfc, bfc, (float*)d_out, tw.C, tw.out_off);
    }
}